// FineMatching_73607149519183
// MI455X (gfx1250) — compile-verified
//
#include <hip/hip_runtime.h>
#include <math.h>

typedef __attribute__((ext_vector_type(2))) float v2f;
typedef __attribute__((ext_vector_type(4))) float v4f;
typedef __attribute__((ext_vector_type(8))) float v8f;

#define WW   64    // 8x8 fine window (l dimension)
#define CDIM 64    // channels
#define RDIM 100   // 10x10 window (r dimension)
#define NP   112   // r padded to 7 tiles of 16

__global__ __launch_bounds__(256) void fine_matching_kernel(
    const float* __restrict__ feat0,   // (M, 64, 64)
    const float* __restrict__ feat1,   // (M, 100, 64)
    const float* __restrict__ mk0,     // (M, 2)
    const float* __restrict__ mk1,     // (M, 2)
    const int*   __restrict__ hw0_i_p, // scalar
    const int*   __restrict__ hw0_f_p, // scalar
    float* __restrict__ out, int M)
{
    __shared__ float sF0[WW * CDIM];    // 16 KB
    __shared__ float sF1[NP * CDIM];    // 28 KB (rows 100..111 zero)
    __shared__ float sConf[WW * NP];    // 28 KB
    __shared__ float rowMax[WW], rowSum[WW];
    __shared__ float colMax[RDIM], colSum[RDIM];
    __shared__ float redV[256];
    __shared__ int   redI[256];
    __shared__ float sWin[9];
    __shared__ int   sIdx;

    const int m    = blockIdx.x;
    const int tid  = threadIdx.x;
    const int lane = tid & 31;
    const int half = lane >> 4;   // 0: K pair {0,1}, 1: K pair {2,3} of each K4 step
    const int ln15 = lane & 15;
    const int wave = tid >> 5;

    const float* f0 = feat0 + (size_t)m * (WW * CDIM);
    const float* f1 = feat1 + (size_t)m * (RDIM * CDIM);

    // ---- Stage A/B into LDS (float4 vectorized) ----
    {
        const v4f* src0 = (const v4f*)f0;
        v4f* dst0 = (v4f*)sF0;
        #pragma unroll
        for (int i = 0; i < 4; ++i) dst0[tid + 256 * i] = src0[tid + 256 * i];

        const v4f* src1 = (const v4f*)f1;
        v4f* dst1 = (v4f*)sF1;
        for (int i = tid; i < (RDIM * CDIM) / 4; i += 256) dst1[i] = src1[i];
        v4f z = {0.f, 0.f, 0.f, 0.f};
        if (tid < ((NP - RDIM) * CDIM) / 4) dst1[(RDIM * CDIM) / 4 + tid] = z;
    }
    __syncthreads();

    // ---- conf_f = (1/64) * f0[:, :56] @ f1[:, :56]^T via V_WMMA_F32_16X16X4_F32 ----
    // 28 tiles (4 row-tiles x 7 col-tiles), 8 waves, 14 K-steps each (K = 0..55).
    for (int t = wave; t < 28; t += 8) {
        const int mt = t / 7, nt = t % 7;
        const int arow = mt * 16 + ln15;   // A: lane<16 -> K pair {k0,k0+1}, lane>=16 -> {k0+2,k0+3}
        const int bcol = nt * 16 + ln15;   // B: same half-split over K
        v8f acc = {};
        #pragma unroll
        for (int kk = 0; kk < 14; ++kk) {
            const int k0 = kk * 4 + half * 2;
            v2f a = *(const v2f*)&sF0[arow * CDIM + k0];
            v2f b = *(const v2f*)&sF1[bcol * CDIM + k0];
            acc = __builtin_amdgcn_wmma_f32_16x16x4_f32(
                false, a, false, b, (short)0, acc, false, false);
        }
        #pragma unroll
        for (int vg = 0; vg < 8; ++vg) {
            const int rr = mt * 16 + vg + 8 * half;  // D layout: VGPR v holds rows v and v+8
            sConf[rr * NP + bcol] = acc[vg] * (1.0f / 64.0f);
        }
    }
    __syncthreads();

    // ---- dual-softmax statistics ----
    if (tid < WW) {                      // per-row (axis=2: over r=0..99)
        float mx = -INFINITY;
        for (int r = 0; r < RDIM; ++r) mx = fmaxf(mx, sConf[tid * NP + r]);
        float s = 0.f;
        for (int r = 0; r < RDIM; ++r) s += expf(sConf[tid * NP + r] - mx);
        rowMax[tid] = mx; rowSum[tid] = s;
    } else if (tid < WW + RDIM) {        // per-col (axis=1: over l=0..63)
        const int r = tid - WW;
        float mx = -INFINITY;
        for (int l = 0; l < WW; ++l) mx = fmaxf(mx, sConf[l * NP + r]);
        float s = 0.f;
        for (int l = 0; l < WW; ++l) s += expf(sConf[l * NP + r] - mx);
        colMax[r] = mx; colSum[r] = s;
    }
    __syncthreads();

    // ---- sm (cropped interior) + write-out + argmax ----
    float bestv = -INFINITY;
    int   besti = 0x7FFFFFFF;
    float* outSm = out + (size_t)13 * M + (size_t)m * (WW * WW);
    #pragma unroll
    for (int it = 0; it < 16; ++it) {
        const int e  = tid + 256 * it;          // l*64 + (i*8 + j)
        const int l  = e >> 6;
        const int ij = e & 63;
        const int r  = ((ij >> 3) + 1) * 10 + (ij & 7) + 1;   // interior of 10x10
        const float c = sConf[l * NP + r];
        const float v = (expf(c - colMax[r]) / colSum[r]) *
                        (expf(c - rowMax[l]) / rowSum[l]);
        outSm[e] = v;
        if (v > bestv || (v == bestv && e < besti)) { bestv = v; besti = e; }
    }
    redV[tid] = bestv; redI[tid] = besti;
    __syncthreads();
    if (tid == 0) {
        float bv = redV[0]; int bi = redI[0];
        for (int i = 1; i < 256; ++i) {
            const float v = redV[i]; const int ii = redI[i];
            if (v > bv || (v == bv && ii < bi)) { bv = v; bi = ii; }
        }
        sIdx = bi;
    }
    __syncthreads();

    const int idx   = sIdx;
    const int idx_l = idx >> 6;
    const int idx_r = idx & 63;

    // ---- conf_ff 3x3 window: 9 threads, one 8-MAC dot each ----
    if (tid < 9) {
        const int di = tid / 3 - 1, dj = tid % 3 - 1;
        int wi = (idx_r >> 3) + di; if (wi < 0) wi += 10;
        int wj = (idx_r & 7)  + dj; if (wj < 0) wj += 10;
        const int rp = wi * 10 + wj;
        float s = 0.f;
        #pragma unroll
        for (int c = 0; c < 8; ++c)
            s += sF0[idx_l * CDIM + 56 + c] * sF1[rp * CDIM + 56 + c];
        sWin[tid] = s * 0.35355339059327373f;   // 1/sqrt(8)
    }
    __syncthreads();

    // ---- final: probs softmax, expectation, keypoints ----
    if (tid == 0) {
        const float scale = (float)hw0_i_p[0] / (float)hw0_f_p[0];
        float w[9], mx = -INFINITY;
        #pragma unroll
        for (int k = 0; k < 9; ++k) { w[k] = sWin[k] * 0.1f; mx = fmaxf(mx, w[k]); }
        float s = 0.f;
        #pragma unroll
        for (int k = 0; k < 9; ++k) { w[k] = expf(w[k] - mx); s += w[k]; }
        const float inv = 1.f / s;
        float cx = 0.f, cy = 0.f;
        float* outP = out + (size_t)4 * M + (size_t)m * 9;
        #pragma unroll
        for (int k = 0; k < 9; ++k) {
            const float p = w[k] * inv;
            outP[k] = p;
            cx += p * (float)((k % 3) - 1);
            cy += p * (float)((k / 3) - 1);
        }
        const float dlx = (float)(idx_l & 7) - 3.5f;
        const float dly = (float)(idx_l >> 3) - 3.5f;
        out[2 * m + 0] = mk0[2 * m + 0] + dlx * scale;
        out[2 * m + 1] = mk0[2 * m + 1] + dly * scale;

        const float drx = (float)(idx_r & 7) - 3.5f;
        const float dry = (float)(idx_r >> 3) - 3.5f;
        out[(size_t)2 * M + 2 * m + 0] = mk1[2 * m + 0] + (drx + cx) * scale;
        out[(size_t)2 * M + 2 * m + 1] = mk1[2 * m + 1] + (dry + cy) * scale;
    }
}

extern "C" void kernel_launch(void* const* d_in, const int* in_sizes, int n_in,
                              void* d_out, int out_size, void* d_ws, size_t ws_size,
                              hipStream_t stream) {
    const float* feat0 = (const float*)d_in[0];
    const float* feat1 = (const float*)d_in[1];
    const float* mk0   = (const float*)d_in[2];
    const float* mk1   = (const float*)d_in[3];
    const int*   hw0_i = (const int*)d_in[6];
    const int*   hw0_f = (const int*)d_in[7];
    const int M = in_sizes[0] / (WW * CDIM);
    fine_matching_kernel<<<M, 256, 0, stream>>>(feat0, feat1, mk0, mk1,
                                                hw0_i, hw0_f, (float*)d_out, M);
}